// MLPPredictor_3985729651445
// MI455X (gfx1250) — compile-verified
//
#include <hip/hip_runtime.h>
#include <hip/hip_bf16.h>

typedef __attribute__((ext_vector_type(2))) float v2f;
typedef __attribute__((ext_vector_type(8))) float v8f;

#define D_FEAT 128
#define WAVES_PER_BLOCK 8
#define BLOCK_THREADS (WAVES_PER_BLOCK * 32)

// ---------------------------------------------------------------------------
// Kernel 1: per-node projection with V_WMMA_F32_16X16X4_F32 (wave32).
//   P[n][0] = h[n] . W[0][0:128]     (src half, class 0)
//   P[n][1] = h[n] . W[1][0:128]     (src half, class 1)
//   P[n][2] = h[n] . W[0][128:256]   (dst half, class 0)
//   P[n][3] = h[n] . W[1][128:256]   (dst half, class 1)
// One wave handles a 16-node tile; 32 accumulating WMMAs over K=128.
// A layout (16x4 f32): lane(m,half) holds {K=k0+2*half, K=k0+2*half+1}.
// B layout (4x16 f32): column n = m; columns 4..15 read a zeroed LDS pad,
// so every lane issues the same unconditional ds_load_b64 (EXEC stays all-1s,
// no saveexec divergence in the hot loop).
// D layout (16x16 f32): VGPR r on lane(m,half) = D[M=r+8*half][N=m].
// ---------------------------------------------------------------------------
__global__ __launch_bounds__(BLOCK_THREADS) void node_project_wmma(
    const float* __restrict__ h, const float* __restrict__ W,
    float* __restrict__ P, int n_tiles)
{
    // 512 floats of W + 128-float zero pad for dead B columns
    __shared__ __align__(16) float sW[512 + 128];
    for (int i = threadIdx.x; i < 640; i += BLOCK_THREADS)
        sW[i] = (i < 512) ? W[i] : 0.0f;
    __syncthreads();

    const int wave = threadIdx.x >> 5;
    const int tile = blockIdx.x * WAVES_PER_BLOCK + wave;
    if (tile >= n_tiles) return;           // wave-uniform: EXEC all-1s below

    const int lane = threadIdx.x & 31;
    const int half = lane >> 4;            // 0: K pair {0,1}, 1: K pair {2,3}
    const int m    = lane & 15;            // node row within tile / B column

    const float2* hrow =
        reinterpret_cast<const float2*>(h + (size_t)(tile * 16 + m) * D_FEAT);

    // B column m -> slice of W (live columns 0..3), dead columns -> zero pad
    const int wbase = (m < 4) ? ((m & 1) * 256 + ((m >> 1) & 1) * 128) : 512;
    const float2* wrow = reinterpret_cast<const float2*>(sW) + (wbase >> 1);

    v8f c = {};
#pragma unroll
    for (int k0 = 0; k0 < D_FEAT; k0 += 4) {
        float2 av = hrow[(k0 >> 1) + half];   // h[node][k0+2h], [k0+2h+1]
        float2 bv = wrow[(k0 >> 1) + half];   // B[k0+2h][m],    B[k0+2h+1][m]
        v2f a; a.x = av.x; a.y = av.y;
        v2f b; b.x = bv.x; b.y = bv.y;
        // 8 args: (neg_a, A, neg_b, B, c_mod, C, reuse_a, reuse_b)
        c = __builtin_amdgcn_wmma_f32_16x16x4_f32(
                false, a, false, b, (short)0, c, false, false);
    }

    if (m < 4) {
#pragma unroll
        for (int r = 0; r < 8; ++r) {
            int node = tile * 16 + r + 8 * half;
            P[(size_t)node * 4 + m] = c[r];
        }
    }
}

// ---------------------------------------------------------------------------
// Kernel 2: per-edge combine. P is 800 KB -> L2/L1 resident gathers.
//   out[e][c] = P[src[e]][c] + P[dst[e]][2+c] + b[c]
// ---------------------------------------------------------------------------
__global__ __launch_bounds__(256) void edge_score_kernel(
    const int* __restrict__ src, const int* __restrict__ dst,
    const float* __restrict__ P, const float* __restrict__ bvec,
    float* __restrict__ out, int E)
{
    int e = blockIdx.x * blockDim.x + threadIdx.x;
    if (e >= E) return;
    int s = src[e];
    int d = dst[e];
    float2 ps = *reinterpret_cast<const float2*>(P + (size_t)s * 4);      // proj 0,1
    float2 pd = *reinterpret_cast<const float2*>(P + (size_t)d * 4 + 2);  // proj 2,3
    float2 o;
    o.x = ps.x + pd.x + bvec[0];
    o.y = ps.y + pd.y + bvec[1];
    *reinterpret_cast<float2*>(out + (size_t)e * 2) = o;
}

extern "C" void kernel_launch(void* const* d_in, const int* in_sizes, int n_in,
                              void* d_out, int out_size, void* d_ws, size_t ws_size,
                              hipStream_t stream)
{
    const float* h   = (const float*)d_in[0];
    const int*   src = (const int*)d_in[1];
    const int*   dst = (const int*)d_in[2];
    const float* W   = (const float*)d_in[3];
    const float* b   = (const float*)d_in[4];
    float*       out = (float*)d_out;
    float*       P   = (float*)d_ws;       // n_nodes * 4 floats (800 KB)

    const int n_nodes = in_sizes[0] / D_FEAT;
    const int E       = in_sizes[1];
    const int n_tiles = (n_nodes + 15) / 16;   // 50000/16 = 3125, exact

    const int grid1 = (n_tiles + WAVES_PER_BLOCK - 1) / WAVES_PER_BLOCK;
    node_project_wmma<<<grid1, BLOCK_THREADS, 0, stream>>>(h, W, P, n_tiles);

    const int threads = 256;
    const int grid2 = (E + threads - 1) / threads;
    edge_score_kernel<<<grid2, threads, 0, stream>>>(src, dst, P, b, out, E);
}